// GCN_44702019616962
// MI455X (gfx1250) — compile-verified
//
#include <hip/hip_runtime.h>
#include <hip/hip_bf16.h>
#include <math.h>

// ---------------------------------------------------------------------------
// GAT layer for MI455X (gfx1250, wave32).
//   K0 init      : emax keys = encode(-inf), denom = 0, num = 0
//   K1 scores    : per-node s_src = f . a_src, s_dst = f . a_dst (wave reduce)
//   K2 edge max  : atomicMax of monotone-int-encoded leaky_relu score per dst
//   K3 decode    : emax float, -inf -> 0
//   K4a edge ex  : ex[e] = exp(e - emax[dst]); atomicAdd denom
//   K4b scatter  : 16 lanes/edge, float4 gather, atomicAdd into num
//   K5 hm        : hm = num/denom (in place over num)
//   K6 gemm      : out = relu([feature | hm] @ lin_w^T + lin_b) via
//                  v_wmma_f32_16x16x4_f32, B staged in LDS (transposed+padded)
// ---------------------------------------------------------------------------

#define IN_FEATS   64
#define CAT_FEATS  128
#define OUT_FEATS  64
#define NEG_SLOPE  0.01f

typedef __attribute__((ext_vector_type(2))) float v2f;
typedef __attribute__((ext_vector_type(8))) float v8f;

// monotone mapping float -> signed int so atomicMax works for any sign
__device__ __forceinline__ int enc_f32(float f) {
    int i = __float_as_int(f);
    return (i >= 0) ? i : (i ^ 0x7FFFFFFF);
}
__device__ __forceinline__ float dec_f32(int k) {
    return __int_as_float((k >= 0) ? k : (k ^ 0x7FFFFFFF));
}

#define ENC_NEG_INF ((int)0x807FFFFF) // enc_f32(-inf)

// ---------------- K0: init scratch ----------------
__global__ void gat_init(int* __restrict__ emax_enc, float* __restrict__ denom,
                         float* __restrict__ num, int n_nodes) {
    int i = blockIdx.x * blockDim.x + threadIdx.x;
    int total = n_nodes * IN_FEATS;
    if (i < total) num[i] = 0.0f;
    if (i < n_nodes) { emax_enc[i] = ENC_NEG_INF; denom[i] = 0.0f; }
}

// ---------------- K1: per-node attention scores ----------------
// one wave (32 lanes) per node; each lane covers 2 of the 64 features
__global__ __launch_bounds__(256) void gat_node_scores(
    const float* __restrict__ feat, const float* __restrict__ attn_w,
    float* __restrict__ s_src, float* __restrict__ s_dst, int n_nodes) {
    const int lane = threadIdx.x & 31;
    const int wave = threadIdx.x >> 5;
    const int node = blockIdx.x * 8 + wave;
    if (node >= n_nodes) return;
    const float f0 = feat[(size_t)node * IN_FEATS + lane];
    const float f1 = feat[(size_t)node * IN_FEATS + 32 + lane];
    float ps = f0 * attn_w[lane]      + f1 * attn_w[32 + lane];
    float pd = f0 * attn_w[64 + lane] + f1 * attn_w[96 + lane];
    #pragma unroll
    for (int off = 16; off > 0; off >>= 1) {
        ps += __shfl_xor(ps, off, 32);
        pd += __shfl_xor(pd, off, 32);
    }
    if (lane == 0) { s_src[node] = ps; s_dst[node] = pd; }
}

// ---------------- K2: per-edge segment max ----------------
__global__ void gat_edge_max(const int* __restrict__ src, const int* __restrict__ dst,
                             const float* __restrict__ s_src, const float* __restrict__ s_dst,
                             int* __restrict__ emax_enc, int n_edges) {
    int e = blockIdx.x * blockDim.x + threadIdx.x;
    if (e >= n_edges) return;
    const int s = src[e], d = dst[e];
    float ev = s_src[s] + s_dst[d];
    ev = (ev < 0.0f) ? ev * NEG_SLOPE : ev;
    atomicMax(emax_enc + d, enc_f32(ev));
}

// ---------------- K3: decode emax, -inf -> 0 ----------------
__global__ void gat_node_fmax(const int* __restrict__ emax_enc,
                              float* __restrict__ emax, int n_nodes) {
    int i = blockIdx.x * blockDim.x + threadIdx.x;
    if (i >= n_nodes) return;
    float f = dec_f32(emax_enc[i]);
    if (!__builtin_isfinite(f)) f = 0.0f;
    emax[i] = f;
}

// ---------------- K4a: per-edge softmax numerator + denom ----------------
__global__ void gat_edge_ex(const int* __restrict__ src, const int* __restrict__ dst,
                            const float* __restrict__ s_src, const float* __restrict__ s_dst,
                            const float* __restrict__ emax,
                            float* __restrict__ exv, float* __restrict__ denom,
                            int n_edges) {
    int e = blockIdx.x * blockDim.x + threadIdx.x;
    if (e >= n_edges) return;
    const int s = src[e], d = dst[e];
    float ev = s_src[s] + s_dst[d];
    ev = (ev < 0.0f) ? ev * NEG_SLOPE : ev;
    const float x = expf(ev - emax[d]);
    exv[e] = x;
    atomicAdd(denom + d, x);
}

// ---------------- K4b: 64-wide weighted scatter ----------------
// 16 lanes per edge; each lane handles 4 contiguous features (float4 gather)
__global__ void gat_edge_scatter(const float* __restrict__ feat,
                                 const int* __restrict__ src, const int* __restrict__ dst,
                                 const float* __restrict__ exv,
                                 float* __restrict__ num, int n_edges) {
    const int t = blockIdx.x * blockDim.x + threadIdx.x;
    const int e = t >> 4;
    const int part = t & 15;
    if (e >= n_edges) return;
    const int s = src[e], d = dst[e];
    const float ex = exv[e];
    const float4 f = *(const float4*)(feat + (size_t)s * IN_FEATS + part * 4);
    float* np = num + (size_t)d * IN_FEATS + part * 4;
    atomicAdd(np + 0, ex * f.x);
    atomicAdd(np + 1, ex * f.y);
    atomicAdd(np + 2, ex * f.z);
    atomicAdd(np + 3, ex * f.w);
}

// ---------------- K5: hm = num / denom (in place) ----------------
__global__ void gat_hm_finalize(float* __restrict__ num, const float* __restrict__ denom,
                                int total) {
    int i = blockIdx.x * blockDim.x + threadIdx.x;
    if (i >= total) return;
    const float dn = denom[i >> 6];
    const float v = num[i];
    num[i] = (dn > 0.0f) ? (v / dn) : 0.0f;
}

// ---------------- K6: WMMA GEMM out = relu([F|hm] @ W^T + b) ----------------
// 4 waves per block; wave nt computes a 16x16 tile: rows [blockIdx*16, +16),
// output columns [nt*16, +16). K = 128 in steps of 4 via v_wmma_f32_16x16x4_f32.
// B (lin_w, 64x128 = 32 KB) is staged once per block into LDS, transposed with
// a +1 row pad so fill stores and inner-loop loads are bank-conflict-free.
#define BSH_STRIDE (OUT_FEATS + 1)   // 65 floats per k-row

__global__ __launch_bounds__(128) void gat_gemm_wmma(
    const float* __restrict__ feat, const float* __restrict__ hm,
    const float* __restrict__ lin_w, const float* __restrict__ lin_b,
    float* __restrict__ out, int n_nodes) {
    __shared__ float bsh[CAT_FEATS * BSH_STRIDE];   // bsh[k*65 + o] = W[o][k]

    // cooperative transpose fill: coalesced global reads along k
    for (int idx = threadIdx.x; idx < OUT_FEATS * CAT_FEATS; idx += 128) {
        const int o = idx >> 7;        // /CAT_FEATS
        const int k = idx & (CAT_FEATS - 1);
        bsh[k * BSH_STRIDE + o] = lin_w[idx];
    }
    __syncthreads();

    const int lane = threadIdx.x & 31;
    const int nt   = threadIdx.x >> 5;     // 0..3 -> output-feature tile
    const int row_base = blockIdx.x * 16;
    const int lo  = lane & 15;
    const int hi  = lane >> 4;             // 0: K+{0,1}, 1: K+{2,3}
    const int o   = nt * 16 + lo;          // output feature column

    // clamp instead of predicate: out-of-range rows compute garbage that is
    // never stored; keeps all loads unconditional (no exec juggling).
    const int arow = min(row_base + lo, n_nodes - 1);
    const float* __restrict__ rowf = feat + (size_t)arow * IN_FEATS;
    const float* __restrict__ rowh = hm   + (size_t)arow * IN_FEATS;

    v8f acc = {};
    #pragma unroll
    for (int k = 0; k < CAT_FEATS; k += 4) {
        const int ka = k + hi * 2;
        // A: contiguous pair -> one global_load_b64
        const float2 av = (k < IN_FEATS)
            ? *(const float2*)(rowf + ka)
            : *(const float2*)(rowh + (ka - IN_FEATS));
        v2f a; a.x = av.x; a.y = av.y;
        // B from LDS (conflict-free: lanes vary o at fixed k)
        v2f b;
        b.x = bsh[ka * BSH_STRIDE + o];
        b.y = bsh[(ka + 1) * BSH_STRIDE + o];
        acc = __builtin_amdgcn_wmma_f32_16x16x4_f32(
            /*neg_a=*/false, a, /*neg_b=*/false, b,
            /*c_mod=*/(short)0, acc, /*reuse_a=*/false, /*reuse_b=*/false);
    }

    const float bias = lin_b[o];
    #pragma unroll
    for (int v = 0; v < 8; ++v) {
        const int m = v + hi * 8;          // C/D layout: lanes 16-31 hold M+8
        const int row = row_base + m;
        if (row < n_nodes) {
            float r = acc[v] + bias;
            out[(size_t)row * OUT_FEATS + o] = (r > 0.0f) ? r : 0.0f;
        }
    }
}

// ---------------------------------------------------------------------------
extern "C" void kernel_launch(void* const* d_in, const int* in_sizes, int n_in,
                              void* d_out, int out_size, void* d_ws, size_t ws_size,
                              hipStream_t stream) {
    const float* feature = (const float*)d_in[0];
    const float* attn_w  = (const float*)d_in[1];
    const float* lin_w   = (const float*)d_in[2];
    const float* lin_b   = (const float*)d_in[3];
    const int*   src     = (const int*)d_in[4];
    const int*   dst     = (const int*)d_in[5];
    float* out = (float*)d_out;

    const int n_nodes = in_sizes[0] / IN_FEATS;   // 50000
    const int n_edges = in_sizes[4];              // 1200000

    // workspace layout (~18.7 MB total; d_ws assumed large enough)
    float* s_src    = (float*)d_ws;
    float* s_dst    = s_src + n_nodes;
    int*   emax_enc = (int*)(s_dst + n_nodes);
    float* emax     = (float*)(emax_enc + n_nodes);
    float* denom    = emax + n_nodes;
    float* num      = denom + n_nodes;            // n_nodes*64; becomes hm
    float* exv      = num + (size_t)n_nodes * IN_FEATS; // n_edges

    const int nf = n_nodes * IN_FEATS;

    gat_init<<<(nf + 255) / 256, 256, 0, stream>>>(emax_enc, denom, num, n_nodes);

    gat_node_scores<<<(n_nodes + 7) / 8, 256, 0, stream>>>(
        feature, attn_w, s_src, s_dst, n_nodes);

    gat_edge_max<<<(n_edges + 255) / 256, 256, 0, stream>>>(
        src, dst, s_src, s_dst, emax_enc, n_edges);

    gat_node_fmax<<<(n_nodes + 255) / 256, 256, 0, stream>>>(
        emax_enc, emax, n_nodes);

    gat_edge_ex<<<(n_edges + 255) / 256, 256, 0, stream>>>(
        src, dst, s_src, s_dst, emax, exv, denom, n_edges);

    {
        const long long threads = (long long)n_edges * 16;
        gat_edge_scatter<<<(int)((threads + 255) / 256), 256, 0, stream>>>(
            feature, src, dst, exv, num, n_edges);
    }

    gat_hm_finalize<<<(nf + 255) / 256, 256, 0, stream>>>(num, denom, nf);

    gat_gemm_wmma<<<(n_nodes + 15) / 16, 128, 0, stream>>>(
        feature, num, lin_w, lin_b, out, n_nodes);
}